// Model_88957362634936
// MI455X (gfx1250) — compile-verified
//
#include <hip/hip_runtime.h>
#include <math.h>

typedef __attribute__((ext_vector_type(2))) float v2f;
typedef __attribute__((ext_vector_type(8))) float v8f;

#define HPI 1.5707963267948966f

// ---------------------------------------------------------------- zero scratch
__global__ void zero_kernel(float* __restrict__ p, size_t n) {
    size_t i = (size_t)blockIdx.x * blockDim.x + threadIdx.x;
    if (i < n) p[i] = 0.0f;
}

// ------------------------------------------------- pose -> Rodrigues -> FK -> A
__global__ void fk_kernel(const float* __restrict__ joints,
                          const float* __restrict__ j0p, const float* __restrict__ j2p,
                          const float* __restrict__ j3p, const float* __restrict__ j4p,
                          const float* __restrict__ j5p,
                          float* __restrict__ Amat, float* __restrict__ lapSlot) {
    if (threadIdx.x != 0 || blockIdx.x != 0) return;
    *lapSlot = 0.0f;

    const int PAR[23] = {-1,0,1,1,3,4,5,4,7,4,9,1,11,12,13,12,15,12,17,0,19,0,21};
    float pose[23][3];
    for (int j = 0; j < 23; ++j)
        for (int c = 0; c < 3; ++c) pose[j][c] = 0.0f;
    for (int c = 0; c < 3; ++c) {
        pose[0][c]  = j0p[c];
        pose[3][c]  = HPI * tanhf(j2p[c]);
        pose[4][c]  = HPI * tanhf(j3p[c]);
        pose[11][c] = HPI * tanhf(j4p[c]);
        pose[12][c] = HPI * tanhf(j5p[c]);
    }

    float chain[23][4][4];
    for (int j = 0; j < 23; ++j) {
        float ax = pose[j][0], ay = pose[j][1], az = pose[j][2];
        float ang = sqrtf(ax*ax + ay*ay + az*az) + 1e-8f;
        float ux = ax / ang, uy = ay / ang, uz = az / ang;
        float s = sinf(ang), cm = 1.0f - cosf(ang);
        float K[3][3] = {{0.f,-uz, uy},{ uz,0.f,-ux},{-uy, ux,0.f}};
        float K2[3][3];
        for (int a = 0; a < 3; ++a)
            for (int b = 0; b < 3; ++b) {
                float acc = 0.f;
                for (int k = 0; k < 3; ++k) acc += K[a][k]*K[k][b];
                K2[a][b] = acc;
            }
        float R[3][3];
        for (int a = 0; a < 3; ++a)
            for (int b = 0; b < 3; ++b)
                R[a][b] = ((a==b)?1.0f:0.0f) + s*K[a][b] + cm*K2[a][b];

        float rel[3];
        if (j == 0) { rel[0]=joints[0]; rel[1]=joints[1]; rel[2]=joints[2]; }
        else {
            int p = PAR[j];
            for (int c = 0; c < 3; ++c) rel[c] = joints[3*j+c] - joints[3*p+c];
        }
        float M[4][4];
        for (int a = 0; a < 3; ++a) {
            for (int b = 0; b < 3; ++b) M[a][b] = R[a][b];
            M[a][3] = rel[a];
        }
        M[3][0]=0.f; M[3][1]=0.f; M[3][2]=0.f; M[3][3]=1.f;

        if (j == 0) {
            for (int a = 0; a < 4; ++a) for (int b = 0; b < 4; ++b) chain[0][a][b] = M[a][b];
        } else {
            int p = PAR[j];
            for (int a = 0; a < 4; ++a)
                for (int b = 0; b < 4; ++b) {
                    float acc = 0.f;
                    for (int k = 0; k < 4; ++k) acc += chain[p][a][k] * M[k][b];
                    chain[j][a][b] = acc;
                }
        }
    }

    for (int j = 0; j < 23; ++j) {
        float jx = joints[3*j], jy = joints[3*j+1], jz = joints[3*j+2];
        for (int a = 0; a < 3; ++a) {
            float corr = chain[j][a][0]*jx + chain[j][a][1]*jy + chain[j][a][2]*jz;
            Amat[j*16 + a*4 + 0] = chain[j][a][0];
            Amat[j*16 + a*4 + 1] = chain[j][a][1];
            Amat[j*16 + a*4 + 2] = chain[j][a][2];
            Amat[j*16 + a*4 + 3] = chain[j][a][3] - corr;
        }
        for (int n = 12; n < 16; ++n) Amat[j*16 + n] = 0.0f;
    }
    for (int n = 0; n < 16; ++n) Amat[23*16 + n] = 0.0f;   // K-padding row
}

// ------------------------------------- skinning GEMM via V_WMMA_F32_16X16X4_F32
// Block = 8 waves = 128 vertices. Skin block is a contiguous 128*JT-float run:
// stage it coalesced into LDS (row stride 25 dwords -> bank-conflict-free),
// then each wave runs KT fully-unrolled WMMA steps with straight-line ds reads.
template<int JT>
__global__ void __launch_bounds__(256)
skin_kernel_t(const float* __restrict__ skin, const float* __restrict__ vin,
              const float* __restrict__ Amat, const float* __restrict__ disp,
              const float* __restrict__ rdis, float* __restrict__ vout, int V) {
    constexpr int KT = (JT + 3) / 4;   // 6 for JT=23
    constexpr int KP = KT * 4;         // 24
    constexpr int SW = KP + 1;         // 25: coprime with 64 banks

    __shared__ float sk[128][SW];
    __shared__ float sA[KP * 16];
    __shared__ float sC[8][16][17];

    int tid = threadIdx.x;
    int blockbase = blockIdx.x * 128;

    // stage transform table (1.5 KB)
    for (int i = tid; i < KP * 16; i += 256) sA[i] = Amat[i];

    // stage skin weights: perfectly coalesced global reads of the contiguous block
    int rows = V - blockbase; if (rows > 128) rows = 128;
    int avail = rows * JT;
    const float* sbase = skin + (size_t)blockbase * JT;
    for (int i = tid; i < 128 * JT; i += 256) {
        float w = (i < avail) ? sbase[i] : 0.0f;
        int row = i / JT;
        int col = i - row * JT;
        sk[row][col] = w;
    }
    // zero the K padding columns
    for (int r = tid; r < 128; r += 256) {
        #pragma unroll
        for (int c = JT; c < KP; ++c) sk[r][c] = 0.0f;
    }

    // prefetch next block's skin region while this block computes
    if (blockbase + 256 <= V) {
        __builtin_prefetch(sbase + (size_t)128 * JT + tid * 11, 0, 1);
    }
    __syncthreads();

    int lane = tid & 31;
    int wave = tid >> 5;
    int half = lane >> 4;
    int col  = lane & 15;
    int srow = wave * 16 + col;        // vertex row within block tile

    v8f c = {0.f,0.f,0.f,0.f,0.f,0.f,0.f,0.f};
    #pragma unroll
    for (int t = 0; t < KT; ++t) {
        int k0 = 4*t + half*2;
        v2f a, b;
        a.x = sk[srow][k0];
        a.y = sk[srow][k0 + 1];
        b.x = sA[(k0    ) * 16 + col];
        b.y = sA[(k0 + 1) * 16 + col];
        c = __builtin_amdgcn_wmma_f32_16x16x4_f32(
                false, a, false, b, (short)0, c, false, false);
    }
    #pragma unroll
    for (int r = 0; r < 8; ++r)
        sC[wave][r + half*8][col] = c[r];
    __syncthreads();

    if (lane < 16) {
        int v = blockbase + wave*16 + lane;
        if (v < V) {
            float x = vin[3*v], y = vin[3*v+1], z = vin[3*v+2];
            const float* T = &sC[wave][lane][0];
            float a0 = rdis[0] + disp[0];
            float a1 = rdis[1] + disp[1];
            float a2 = rdis[2] + disp[2];
            vout[3*v+0] = T[0]*x + T[1]*y + T[2]*z  + T[3]  + a0;
            vout[3*v+1] = T[4]*x + T[5]*y + T[6]*z  + T[7]  + a1;
            vout[3*v+2] = T[8]*x + T[9]*y + T[10]*z + T[11] + a2;
        }
    }
}

// generic fallback (runtime J) — same algorithm, scalar loop, only used if J != 23
__global__ void __launch_bounds__(256)
skin_kernel_gen(const float* __restrict__ skin, const float* __restrict__ vin,
                const float* __restrict__ Amat, const float* __restrict__ disp,
                const float* __restrict__ rdis, float* __restrict__ vout,
                int V, int J) {
    int v = blockIdx.x * blockDim.x + threadIdx.x;
    if (v >= V) return;
    float T[12];
    #pragma unroll
    for (int n = 0; n < 12; ++n) T[n] = 0.0f;
    for (int j = 0; j < J; ++j) {
        float w = skin[(size_t)v * J + j];
        #pragma unroll
        for (int n = 0; n < 12; ++n) T[n] += w * Amat[j*16 + n];
    }
    float x = vin[3*v], y = vin[3*v+1], z = vin[3*v+2];
    vout[3*v+0] = T[0]*x + T[1]*y + T[2]*z  + T[3]  + rdis[0] + disp[0];
    vout[3*v+1] = T[4]*x + T[5]*y + T[6]*z  + T[7]  + rdis[1] + disp[1];
    vout[3*v+2] = T[8]*x + T[9]*y + T[10]*z + T[11] + rdis[2] + disp[2];
}

// -------------------------------------------- neighbor-sum scatter (Laplacian)
__global__ void face_kernel(const int* __restrict__ faces, const float* __restrict__ verts,
                            float* __restrict__ nsum, float* __restrict__ deg, int F) {
    int f = blockIdx.x * blockDim.x + threadIdx.x;
    if (f >= F) return;
    int i0 = faces[3*f], i1 = faces[3*f+1], i2 = faces[3*f+2];
    float a0=verts[3*i0], a1=verts[3*i0+1], a2=verts[3*i0+2];
    float b0=verts[3*i1], b1=verts[3*i1+1], b2=verts[3*i1+2];
    float c0=verts[3*i2], c1=verts[3*i2+1], c2=verts[3*i2+2];
    atomicAdd(&nsum[3*i0+0], b0+c0); atomicAdd(&nsum[3*i0+1], b1+c1); atomicAdd(&nsum[3*i0+2], b2+c2);
    atomicAdd(&nsum[3*i1+0], c0+a0); atomicAdd(&nsum[3*i1+1], c1+a1); atomicAdd(&nsum[3*i1+2], c2+a2);
    atomicAdd(&nsum[3*i2+0], a0+b0); atomicAdd(&nsum[3*i2+1], a1+b1); atomicAdd(&nsum[3*i2+2], a2+b2);
    atomicAdd(&deg[i0], 2.0f); atomicAdd(&deg[i1], 2.0f); atomicAdd(&deg[i2], 2.0f);
}

// ---------------------------- Lv, lap reduction, and B-way output replication
__global__ void __launch_bounds__(256)
lap_kernel(const float* __restrict__ verts, const float* __restrict__ nsum,
           const float* __restrict__ deg, float* __restrict__ out,
           float* __restrict__ lap, int V, int B) {
    int v = blockIdx.x * blockDim.x + threadIdx.x;
    float sq = 0.0f;
    if (v < V) {
        float vx = verts[3*v], vy = verts[3*v+1], vz = verts[3*v+2];
        float d = fmaxf(deg[v], 1.0f);
        float lx = vx - nsum[3*v+0] / d;
        float ly = vy - nsum[3*v+1] / d;
        float lz = vz - nsum[3*v+2] / d;
        sq = lx*lx + ly*ly + lz*lz;
        size_t stride = (size_t)V * 3;
        for (int r = 1; r < B; ++r) {
            out[(size_t)r*stride + 3*v+0] = vx;
            out[(size_t)r*stride + 3*v+1] = vy;
            out[(size_t)r*stride + 3*v+2] = vz;
        }
    }
    __shared__ float red[256];
    red[threadIdx.x] = sq;
    __syncthreads();
    for (int s = 128; s > 0; s >>= 1) {
        if ((int)threadIdx.x < s) red[threadIdx.x] += red[threadIdx.x + s];
        __syncthreads();
    }
    if (threadIdx.x == 0) atomicAdd(lap, red[0]);
}

extern "C" void kernel_launch(void* const* d_in, const int* in_sizes, int n_in,
                              void* d_out, int out_size, void* d_ws, size_t ws_size,
                              hipStream_t stream) {
    const float* vertices = (const float*)d_in[0];
    const float* joints   = (const float*)d_in[1];
    const float* skin     = (const float*)d_in[2];
    const float* j0p      = (const float*)d_in[3];
    const float* j2p      = (const float*)d_in[4];
    const float* j3p      = (const float*)d_in[5];
    const float* j4p      = (const float*)d_in[6];
    const float* j5p      = (const float*)d_in[7];
    const float* disp     = (const float*)d_in[8];
    const float* rdis     = (const float*)d_in[9];
    const int*   faces    = (const int*)d_in[10];

    int V = in_sizes[0] / 3;
    int J = in_sizes[1] / 3;
    int F = in_sizes[10] / 3;
    long long vb3 = (long long)V * 3;
    int B = (int)(((long long)out_size - 1) / vb3);
    if (B < 1) B = 1;

    float* ws   = (float*)d_ws;
    float* Amat = ws;                              // 24x16 padded transforms
    float* nsum = ws + 512;                        // V*3 floats
    float* deg  = ws + 512 + (size_t)V * 3;        // V floats
    float* out  = (float*)d_out;
    float* lap  = out + (size_t)B * V * 3;         // scalar slot

    size_t nz = (size_t)V * 4;
    zero_kernel<<<(unsigned)((nz + 255) / 256), 256, 0, stream>>>(nsum, nz);

    fk_kernel<<<1, 32, 0, stream>>>(joints, j0p, j2p, j3p, j4p, j5p, Amat, lap);

    if (J == 23) {
        int blocks = (V + 127) / 128;
        skin_kernel_t<23><<<blocks, 256, 0, stream>>>(skin, vertices, Amat, disp, rdis, out, V);
    } else {
        skin_kernel_gen<<<(V + 255) / 256, 256, 0, stream>>>(skin, vertices, Amat, disp, rdis, out, V, J);
    }

    face_kernel<<<(F + 255) / 256, 256, 0, stream>>>(faces, out, nsum, deg, F);

    lap_kernel<<<(V + 255) / 256, 256, 0, stream>>>(out, nsum, deg, out, lap, V, B);
}